// TopK_50594714747199
// MI455X (gfx1250) — compile-verified
//
#include <hip/hip_runtime.h>

#define ROWLEN 4096
#define KKEEP  410
#define TPB    256
#define EPT    16   // elements per thread (TPB * EPT == ROWLEN)

typedef float v4f __attribute__((ext_vector_type(4)));

__global__ __launch_bounds__(TPB)
void topk_mask_kernel(const float* __restrict__ x, float* __restrict__ out) {
    __shared__ float    row[ROWLEN];        // 16 KB staged row
    __shared__ unsigned hist[8 * 256];      // per-wave histograms (8 waves)
    __shared__ unsigned hsum[256];          // reduced histogram -> suffix sums
    __shared__ unsigned scanbuf[256];       // tie-rank scan
    __shared__ unsigned sel[2];             // [0]=selected digit, [1]=remaining count

    const unsigned tid  = threadIdx.x;
    const unsigned wave = tid >> 5;         // wave32: 8 waves per block
    const unsigned long long rowBase = (unsigned long long)blockIdx.x * ROWLEN;
    const float* gsrc = x + rowBase;

    // ---------------- Stage row into LDS via async global->LDS DMA ----------------
    // 4 chunks of 4 KB; each lane moves 16 B per chunk (fully coalesced).
    {
        unsigned ldsBase = (unsigned)(unsigned long long)(&row[0]);
#pragma unroll
        for (int c = 0; c < 4; ++c) {
            unsigned byteOff = (unsigned)(c * 4096) + tid * 16u;
            unsigned ldsAddr = ldsBase + byteOff;
            unsigned vOff    = byteOff;
            asm volatile("global_load_async_to_lds_b128 %0, %1, %2"
                         :
                         : "v"(ldsAddr), "v"(vOff), "s"(gsrc)
                         : "memory");
        }
        asm volatile("s_wait_asynccnt 0x0" ::: "memory");
    }
    __syncthreads();

    // ---------------- Pull my 16 contiguous elements into registers ----------------
    float    v[EPT];
    unsigned b[EPT];
#pragma unroll
    for (int i = 0; i < EPT; ++i) {
        v[i] = row[tid * EPT + i];
        b[i] = __float_as_uint(v[i]) & 0x7fffffffu;   // |x| bits: monotonic as u32
    }

    // ---------------- 4-pass MSB-first radix select for the K-th largest ----------------
    unsigned prefix = 0u, pmask = 0u, want = KKEEP;
    for (int d = 3; d >= 0; --d) {
        for (unsigned i = tid; i < 8u * 256u; i += TPB) hist[i] = 0u;
        __syncthreads();

        const unsigned sh = (unsigned)d * 8u;
        unsigned* myh = &hist[wave * 256u];
#pragma unroll
        for (int i = 0; i < EPT; ++i) {
            if ((b[i] & pmask) == prefix)
                atomicAdd(&myh[(b[i] >> sh) & 255u], 1u);   // ds_add_u32
        }
        __syncthreads();

        // reduce the 8 per-wave histograms into hsum
        unsigned s = 0u;
#pragma unroll
        for (int w = 0; w < 8; ++w) s += hist[w * 256u + tid];
        hsum[tid] = s;
        __syncthreads();

        // parallel suffix sum (descending inclusive): hsum[t] = sum_{j>=t} count[j]
#pragma unroll
        for (int off = 1; off < TPB; off <<= 1) {
            unsigned mine = hsum[tid];
            unsigned add  = (tid + (unsigned)off < TPB) ? hsum[tid + off] : 0u;
            __syncthreads();
            hsum[tid] = mine + add;
            __syncthreads();
        }

        // unique bin with suffix >= want and suffix(above) < want holds the want-th largest
        {
            unsigned suff  = hsum[tid];
            unsigned above = (tid < TPB - 1u) ? hsum[tid + 1u] : 0u;
            if (suff >= want && above < want) {
                sel[0] = tid;            // selected digit
                sel[1] = want - above;   // still needed inside this bin
            }
        }
        __syncthreads();

        prefix |= sel[0] << sh;
        pmask  |= 0xFFu << sh;
        want    = sel[1];
        __syncthreads();
    }

    const unsigned T = prefix;   // threshold key; keep first `want` elements equal to it

    // ---------------- Stable tie ranking (index order == jax top_k tie order) ----------------
    unsigned eq = 0u;
#pragma unroll
    for (int i = 0; i < EPT; ++i) eq += (b[i] == T) ? 1u : 0u;
    scanbuf[tid] = eq;
    __syncthreads();
#pragma unroll
    for (int off = 1; off < TPB; off <<= 1) {
        unsigned mine = scanbuf[tid];
        unsigned add  = (tid >= (unsigned)off) ? scanbuf[tid - off] : 0u;
        __syncthreads();
        scanbuf[tid] = mine + add;
        __syncthreads();
    }
    const unsigned base = scanbuf[tid] - eq;   // equals in lower-index threads

    // ---------------- Mask and stream out (nontemporal: written once, never reread) ----------------
    float ov[EPT];
    unsigned r = 0u;
#pragma unroll
    for (int i = 0; i < EPT; ++i) {
        bool keep = (b[i] > T) || ((b[i] == T) && ((base + r) < want));
        if (b[i] == T) ++r;
        ov[i] = keep ? v[i] : 0.0f;
    }

    v4f* dst = (v4f*)(out + rowBase + (unsigned long long)tid * EPT);
#pragma unroll
    for (int q = 0; q < 4; ++q) {
        v4f t4 = { ov[4 * q + 0], ov[4 * q + 1], ov[4 * q + 2], ov[4 * q + 3] };
        __builtin_nontemporal_store(t4, dst + q);
    }
}

extern "C" void kernel_launch(void* const* d_in, const int* in_sizes, int n_in,
                              void* d_out, int out_size, void* d_ws, size_t ws_size,
                              hipStream_t stream) {
    (void)n_in; (void)d_ws; (void)ws_size; (void)out_size;
    const float* x  = (const float*)d_in[0];
    float*       out = (float*)d_out;
    const int n    = in_sizes[0];        // 4 * 4096 * 4096
    const int rows = n / ROWLEN;         // 16384 rows
    topk_mask_kernel<<<rows, TPB, 0, stream>>>(x, out);
}